// CubicalModel_ISM_norm_46746424049888
// MI455X (gfx1250) — compile-verified
//
#include <hip/hip_runtime.h>

// Cubical persistence (dim 1, card 50) of Ip = I(28,28,2) . p(2,1).
// Single workgroup, LDS-resident; wave32 bitset boundary reduction;
// WMMA f32 16x16x4 for the matvec front-end.

#define Hh 28
#define Ww 28
#define AB 55                 // 2*28-1
#define NCELL (AB*AB)         // 3025 cells
#define WORDS 96              // ceil(3025/32)=95, padded to 3 dwords/lane
#define CARDN 50
#define MAXROWS 1536          // max dim-1 births = 1512 edges
#define NTHREADS 1024

typedef float v2f __attribute__((ext_vector_type(2)));
typedef float v8f __attribute__((ext_vector_type(8)));

struct SmemT {
  float          Ip[Hh*Ww];       // 3136 B
  float          F[NCELL];        // 12100 B  filtration value per cell
  float          pers[MAXROWS];   // 6144 B
  unsigned short P[NCELL];        // 6050 B   pixel index realizing F
  unsigned short pos[NCELL];      // 6050 B   rank in filtration order
  unsigned short order[NCELL];    // 6050 B   cell at each rank
  short          low[NCELL];      // 6050 B   low -> column map (-1 = none)
  unsigned char  dms[NCELL];      // 3025 B   cell dimension
  unsigned short rowB[MAXROWS];   // 3072 B   birth pixel
  unsigned short rowD[MAXROWS];   // 3072 B   death pixel
  unsigned short srt[MAXROWS];    // 3072 B   persistence-sorted order
  int            nrows;
};                                // ~58 KB total

__global__ __launch_bounds__(NTHREADS)
void cubical_persistence_kernel(const float* __restrict__ I,
                                const float* __restrict__ p,
                                float* __restrict__ out,
                                unsigned* __restrict__ ws)
{
  __shared__ SmemT sm;
  const int tid  = threadIdx.x;
  const int lane = tid & 31;
  const int wv   = tid >> 5;          // 32 waves

  // ---------------- Phase 0: Ip = I . p via V_WMMA_F32_16X16X4_F32 --------
  // A tile = 16 pixels x K(=2, padded to 4); B = p broadcast to all 16 cols.
  // D column 0..15 all equal the 16 Ip values of the tile.
  const float p0 = p[0], p1 = p[1];
  __builtin_prefetch(I, 0, 3);        // global_prefetch_b8
  for (int t = wv; t < 49; t += 32) { // wave-uniform trip count, EXEC stays full
    const bool hi = (lane >= 16);     // lanes 16..31 hold K=2,3 (zero padding)
    const int  m  = lane & 15;
    const int  pix = t * 16 + m;
    v2f a, b;
    a[0] = hi ? 0.f : I[pix * 2 + 0]; // K=0 (or K=2 -> 0)
    a[1] = hi ? 0.f : I[pix * 2 + 1]; // K=1 (or K=3 -> 0)
    b[0] = hi ? 0.f : p0;
    b[1] = hi ? 0.f : p1;
    v8f c = {0.f,0.f,0.f,0.f,0.f,0.f,0.f,0.f};
    c = __builtin_amdgcn_wmma_f32_16x16x4_f32(
            /*neg_a=*/false, a, /*neg_b=*/false, b,
            /*c_mod=*/(short)0, c, /*reuse_a=*/false, /*reuse_b=*/false);
    if (m == 0) {                     // one lane per half writes M=v(+8), N=0
      const int mb = t * 16 + (hi ? 8 : 0);
      #pragma unroll
      for (int v = 0; v < 8; ++v) sm.Ip[mb + v] = c[v];
    }
  }
  __syncthreads();

  // ---------------- Phase 1: per-cell filtration (F, P, dim) --------------
  for (int cell = tid; cell < NCELL; cell += NTHREADS) {
    const int a = cell / AB, b = cell - a * AB;
    int rs[2]; int nr = 0;
    if (a & 1) { rs[nr++] = (a - 1) >> 1; }
    else { const int r = a >> 1; if (r - 1 >= 0) rs[nr++] = r - 1; if (r < Hh) rs[nr++] = r; }
    int cs[2]; int nc = 0;
    if (b & 1) { cs[nc++] = (b - 1) >> 1; }
    else { const int cc = b >> 1; if (cc - 1 >= 0) cs[nc++] = cc - 1; if (cc < Ww) cs[nc++] = cc; }
    float best = 0.f; int bp = 0; bool first = true;
    for (int ri = 0; ri < nr; ++ri)
      for (int ci = 0; ci < nc; ++ci) {
        const int pix = rs[ri] * Ww + cs[ci];
        const float v = sm.Ip[pix];
        if (first || v < best) { best = v; bp = pix; first = false; }
      }
    sm.F[cell]   = best;
    sm.P[cell]   = (unsigned short)bp;
    sm.dms[cell] = (unsigned char)((a & 1) + (b & 1));
  }
  __syncthreads();

  // ---------------- Phase 2: sort by (F, dim, index) via rank counting ----
  for (int i = tid; i < NCELL; i += NTHREADS) sm.low[i] = -1;
  for (int i = tid; i < NCELL; i += NTHREADS) {
    const float Fi = sm.F[i];
    const int   di = sm.dms[i];
    int r = 0;
    for (int j = 0; j < NCELL; ++j) {
      const float Fj = sm.F[j];
      const int   dj = sm.dms[j];
      const bool lt = (Fj < Fi) || (Fj == Fi && (dj < di || (dj == di && j < i)));
      r += lt ? 1 : 0;
    }
    sm.pos[i]   = (unsigned short)r;
    sm.order[r] = (unsigned short)i;   // ranks are unique: no race
  }
  __syncthreads();

  // ---------------- Phase 3: Z/2 boundary reduction (wave 0 only) ---------
  // Column = 3025-bit set, 3 dwords/lane: word w lives in lane (w&31), slot (w>>5).
  if (wv == 0) {
    volatile short* vlow = sm.low;
    int nrows = 0;
    for (int j = 0; j < NCELL; ++j) {
      const int cell = sm.order[j];
      const int a = cell / AB, b = cell - a * AB;
      unsigned w0 = 0, w1 = 0, w2 = 0;
      int fc[4]; int nf = 0;
      if (a & 1) { fc[nf++] = cell - AB; fc[nf++] = cell + AB; }
      if (b & 1) { fc[nf++] = cell - 1;  fc[nf++] = cell + 1;  }
      for (int q = 0; q < nf; ++q) {
        const int pp = sm.pos[fc[q]];
        const int w  = pp >> 5;
        const unsigned bit = 1u << (pp & 31);
        if ((w & 31) == lane) {
          const int s = w >> 5;
          if (s == 0) w0 ^= bit; else if (s == 1) w1 ^= bit; else w2 ^= bit;
        }
      }
      for (;;) {
        // low(col): per-lane MSB candidate, then wave32 max-reduce
        const int c0 = w0 ? (( lane        << 5) | (31 - __clz((int)w0))) : -1;
        const int c1 = w1 ? (((32 + lane)  << 5) | (31 - __clz((int)w1))) : -1;
        const int c2 = w2 ? (((64 + lane)  << 5) | (31 - __clz((int)w2))) : -1;
        int best = c0 > c1 ? c0 : c1; if (c2 > best) best = c2;
        #pragma unroll
        for (int off = 16; off > 0; off >>= 1) {
          const int o = __shfl_xor(best, off, 32);
          if (o > best) best = o;
        }
        if (best < 0) break;               // column vanished: no pair
        const int l = best;                // uniform across wave
        const int k = vlow[l];
        if (k < 0) {
          if (lane == 0) vlow[l] = (short)j;
          unsigned* cp = ws + (unsigned)j * WORDS;   // persist reduced column
          cp[lane] = w0; cp[32 + lane] = w1; cp[64 + lane] = w2;
          const int bc = sm.order[l];
          if (sm.dms[bc] == 1) {                      // dim-1 pair
            const float pr = sm.F[cell] - sm.F[bc];
            if (pr > 0.f && nrows < MAXROWS) {
              if (lane == 0) {
                sm.rowB[nrows] = sm.P[bc];
                sm.rowD[nrows] = sm.P[cell];
                sm.pers[nrows] = pr;
              }
              nrows++;                                // uniform
            }
          }
          asm volatile("s_wait_dscnt 0" ::: "memory");
          break;
        } else {
          const unsigned* cp = ws + (unsigned)k * WORDS;
          w0 ^= cp[lane]; w1 ^= cp[32 + lane]; w2 ^= cp[64 + lane];
        }
      }
    }
    if (lane == 0) sm.nrows = nrows;
  }
  __syncthreads();

  // ---------------- Phase 4: sort pairs by persistence descending ---------
  // Matches np.argsort(stable asc)[::-1]: ties broken by larger index first.
  const int m = sm.nrows;
  for (int k2 = tid; k2 < m; k2 += NTHREADS) {
    const float pk = sm.pers[k2];
    int r = 0;
    for (int q = 0; q < m; ++q) {
      const float pq = sm.pers[q];
      const bool before = (pq > pk) || (pq == pk && q > k2);
      r += before ? 1 : 0;
    }
    sm.srt[r] = (unsigned short)k2;
  }
  __syncthreads();

  // ---------------- Phase 5: emit diagram (card=50, pad gathers Ip[0,0]) --
  if (tid < 2 * CARDN) {
    const int k2 = tid >> 1;
    float val;
    if (k2 < m) {
      const int s   = sm.srt[k2];
      const int pix = (tid & 1) ? sm.rowD[s] : sm.rowB[s];
      val = sm.Ip[pix];
    } else {
      val = sm.Ip[0];   // reference pads indices with 0 then gathers
    }
    out[tid] = val;
  }
}

extern "C" void kernel_launch(void* const* d_in, const int* in_sizes, int n_in,
                              void* d_out, int out_size, void* d_ws, size_t ws_size,
                              hipStream_t stream) {
  const float* I = (const float*)d_in[0];   // (28,28,2) float32
  const float* p = (const float*)d_in[1];   // (2,1)    float32
  float* out = (float*)d_out;               // (50,2)   float32 -> 100 elems
  unsigned* ws = (unsigned*)d_ws;           // needs 3025*96*4 = 1,161,600 B
  (void)in_sizes; (void)n_in; (void)out_size; (void)ws_size;
  hipLaunchKernelGGL(cubical_persistence_kernel, dim3(1), dim3(NTHREADS), 0, stream,
                     I, p, out, ws);
}